// GraphAttentionLayer_59107339927921
// MI455X (gfx1250) — compile-verified
//
#include <hip/hip_runtime.h>
#include <hip/hip_bf16.h>
#include <math.h>

#define N_NODES   50000
#define N_EDGES   800000
#define IN_CH     256
#define HEADS     4
#define DHEAD     64
#define HD        (HEADS * DHEAD)   // 256
#define NEG_SLOPE 0.2f

typedef __attribute__((ext_vector_type(2))) float v2f;
typedef __attribute__((ext_vector_type(8))) float v8f;

// ---------------------------------------------------------------------------
// W_red[i,d] = 0.25 * sum_h W_res[i, h*64 + d]   (residual folded with head-mean)
// ---------------------------------------------------------------------------
__global__ void gat_wres_reduce(const float* __restrict__ Wres,
                                float* __restrict__ Wred) {
    int i = blockIdx.x * blockDim.x + threadIdx.x;       // 256*64 = 16384
    if (i >= IN_CH * DHEAD) return;
    int r = i / DHEAD, d = i % DHEAD;
    const float* p = Wres + (size_t)r * HD + d;
    Wred[i] = 0.25f * (p[0] + p[64] + p[128] + p[192]);
}

// ---------------------------------------------------------------------------
// fp32 WMMA GEMM: C[M x NCOLS] = A[M x 256] * B[256 x NCOLS]
//   - block = 256 threads = 8 waves; each wave owns a 16-row x 64-col strip
//     (4 accumulators -> 4 independent v_wmma per K-step, 4x A reuse)
//   - B staged in LDS with K-row PAIRS interleaved by column:
//         Blds[j*PITCH + c*2 + {0,1}] = B[2j][c], B[2j+1][c]
//     so each WMMA B operand (rows kl+2h, kl+1+2h at one column) is a single
//     contiguous ds_load_b64 straight into the operand VGPR pair (no movs).
//   - PITCH = 160 floats: 160 mod 64 = 32 -> half=0 lanes (even j) and half=1
//     lanes (odd j) sit 32 banks apart; columns cover the 16 even banks and
//     the b64 partner dwords the odd banks => all 64 banks distinct.
//   - row-tile index CLAMPED (not branched): EXEC all-ones at every WMMA,
//     all threads reach the barriers; overflow waves rewrite the last tile
//     with identical values (benign).
// ---------------------------------------------------------------------------
template<int NCOLS>
__global__ __launch_bounds__(256) void gat_gemm_f32_lds(
        const float* __restrict__ A,
        const float* __restrict__ B,
        float* __restrict__ C) {
    constexpr int STRIP   = 64;              // columns per block
    constexpr int KH      = 128;             // K rows staged per phase
    constexpr int PITCH   = 160;             // floats per interleaved pair-row
    constexpr int ROWTILES = N_NODES / 16;   // 3125
    constexpr int BLK_PER_STRIP = (ROWTILES + 7) / 8;   // 391

    __shared__ float Blds[(KH / 2) * PITCH]; // 64 pair-rows * 160 = 40 KB

    const int strip  = blockIdx.x / BLK_PER_STRIP;
    const int rowBlk = blockIdx.x % BLK_PER_STRIP;
    const int n0     = strip * STRIP;

    const int wave = threadIdx.x >> 5;
    int rowTile = rowBlk * 8 + wave;
    if (rowTile >= ROWTILES) rowTile = ROWTILES - 1;     // clamp, no divergence
    const int lane = threadIdx.x & 31;
    const int half = lane >> 4;                          // 0 | 1
    const int lm   = lane & 15;
    const int m0   = rowTile << 4;

    const float* arow  = A + (size_t)(m0 + lm) * IN_CH + 2 * half;
    const float* bbase = Blds + lm * 2;                  // column lm pairs

    v8f acc[4] = {};

    for (int kb = 0; kb < IN_CH; kb += KH) {
        __syncthreads();                                 // LDS reuse fence
        // ---- cooperative stage: interleave B[kb+2j, kb+2j+1][n0+c] ----
        {
            const int t = threadIdx.x;
#pragma unroll
            for (int i = 0; i < (KH / 2) * (STRIP / 2) / 256; ++i) {  // 8 each
                int q  = i * 256 + t;
                int j  = q >> 5;                         // pair-row (0..63)
                int c2 = (q & 31) << 1;                  // column (step 2)
                const float* g = B + (size_t)(kb + 2 * j) * NCOLS + n0 + c2;
                float2 r0 = *(const float2*)(g);         // row 2j,   cols c2,c2+1
                float2 r1 = *(const float2*)(g + NCOLS); // row 2j+1, cols c2,c2+1
                float4 w; w.x = r0.x; w.y = r1.x; w.z = r0.y; w.w = r1.y;
                *(float4*)(&Blds[j * PITCH + c2 * 2]) = w;
            }
        }
        __syncthreads();

        // ---- compute: 32 K-steps of 4, 4 tiles wide ----
        const float* ak = arow + kb;
#pragma unroll 4
        for (int kl = 0; kl < KH; kl += 4) {
            const float2 t2 = *(const float2*)(ak + kl);
            v2f a; a.x = t2.x; a.y = t2.y;
            const float* bk = bbase + ((kl >> 1) + half) * PITCH;
#pragma unroll
            for (int t = 0; t < 4; ++t) {
                v2f b = *(const v2f*)(bk + t * 32);      // one ds_load_b64
                acc[t] = __builtin_amdgcn_wmma_f32_16x16x4_f32(
                    /*neg_a=*/false, a, /*neg_b=*/false, b,
                    /*c_mod=*/(short)0, acc[t],
                    /*reuse_a=*/false, /*reuse_b=*/false);
            }
        }
    }

    // D layout: VGPR j -> row (j + 8*half), col lm
    float* crow = C + (size_t)(m0 + 8 * half) * NCOLS + n0 + lm;
#pragma unroll
    for (int t = 0; t < 4; ++t)
#pragma unroll
        for (int j = 0; j < 8; ++j)
            crow[(size_t)j * NCOLS + t * 16] = acc[t][j];
}

// ---------------------------------------------------------------------------
// el[n,h] = <h[n,h,:], attn_l[h,:]>, er likewise (single pass over h)
// ---------------------------------------------------------------------------
__global__ void gat_logits(const float* __restrict__ hfeat,
                           const float* __restrict__ attn_l,
                           const float* __restrict__ attn_r,
                           float* __restrict__ el,
                           float* __restrict__ er) {
    int idx = blockIdx.x * blockDim.x + threadIdx.x;     // n*HEADS + h
    if (idx >= N_NODES * HEADS) return;
    int n = idx >> 2, hh = idx & 3;
    const float4* hv = (const float4*)(hfeat + (size_t)n * HD + hh * DHEAD);
    const float4* al = (const float4*)(attn_l + hh * DHEAD);
    const float4* ar = (const float4*)(attn_r + hh * DHEAD);
    float sl = 0.f, sr = 0.f;
#pragma unroll
    for (int i = 0; i < DHEAD / 4; ++i) {
        float4 v = hv[i], a = al[i], b = ar[i];
        sl += v.x * a.x + v.y * a.y + v.z * a.z + v.w * a.w;
        sr += v.x * b.x + v.y * b.y + v.z * b.z + v.w * b.w;
    }
    el[idx] = sl;
    er[idx] = sr;
}

__global__ void gat_init(float* __restrict__ m, float* __restrict__ s) {
    int i = blockIdx.x * blockDim.x + threadIdx.x;
    if (i < N_NODES * HEADS) { m[i] = -INFINITY; s[i] = 0.f; }
}

// sign-split float atomic max: positive -> int max, negative -> uint min
__device__ __forceinline__ void atomicMaxF(float* addr, float v) {
    if (v >= 0.f) atomicMax((int*)addr, __float_as_int(v));
    else          atomicMin((unsigned int*)addr, __float_as_uint(v));
}

__device__ __forceinline__ float leaky(float x) {
    return x >= 0.f ? x : NEG_SLOPE * x;
}

__global__ void gat_edge_max(const int* __restrict__ src,
                             const int* __restrict__ dst,
                             const float* __restrict__ el,
                             const float* __restrict__ er,
                             float* __restrict__ m) {
    int i = blockIdx.x * blockDim.x + threadIdx.x;       // e*HEADS + h
    if (i >= N_EDGES * HEADS) return;
    int e = i >> 2, hh = i & 3;
    float sc = leaky(el[src[e] * HEADS + hh] + er[dst[e] * HEADS + hh]);
    atomicMaxF(&m[dst[e] * HEADS + hh], sc);
}

__global__ void gat_edge_sum(const int* __restrict__ src,
                             const int* __restrict__ dst,
                             const float* __restrict__ el,
                             const float* __restrict__ er,
                             const float* __restrict__ m,
                             float* __restrict__ s) {
    int i = blockIdx.x * blockDim.x + threadIdx.x;
    if (i >= N_EDGES * HEADS) return;
    int e = i >> 2, hh = i & 3;
    int v = dst[e] * HEADS + hh;
    float sc = leaky(el[src[e] * HEADS + hh] + er[v]);
    atomicAdd(&s[v], __expf(sc - m[v]));
}

// ---------------------------------------------------------------------------
// Wave-per-edge: recompute the 4 alphas, then out[dst, d] += 0.25 * sum_h
// alpha_h * h[src, h, d] for d = lane, lane+32. h is L2-resident (51MB<192MB).
// ---------------------------------------------------------------------------
__global__ void gat_aggregate(const int* __restrict__ src,
                              const int* __restrict__ dst,
                              const float* __restrict__ el,
                              const float* __restrict__ er,
                              const float* __restrict__ m,
                              const float* __restrict__ s,
                              const float* __restrict__ hfeat,
                              float* __restrict__ out) {
    int wave = (blockIdx.x * blockDim.x + threadIdx.x) >> 5;
    if (wave >= N_EDGES) return;
    int lane = threadIdx.x & 31;
    int u = src[wave], v = dst[wave];

    float alpha[HEADS];
#pragma unroll
    for (int hh = 0; hh < HEADS; ++hh) {
        int vi = v * HEADS + hh;
        float sc = leaky(el[u * HEADS + hh] + er[vi]);
        alpha[hh] = __expf(sc - m[vi]) / s[vi];
    }
    const float* hu = hfeat + (size_t)u * HD;
    float a0 = 0.f, a1 = 0.f;
#pragma unroll
    for (int hh = 0; hh < HEADS; ++hh) {
        a0 += alpha[hh] * hu[hh * DHEAD + lane];
        a1 += alpha[hh] * hu[hh * DHEAD + lane + 32];
    }
    atomicAdd(&out[(size_t)v * DHEAD + lane],      0.25f * a0);
    atomicAdd(&out[(size_t)v * DHEAD + lane + 32], 0.25f * a1);
}

// ---------------------------------------------------------------------------
extern "C" void kernel_launch(void* const* d_in, const int* in_sizes, int n_in,
                              void* d_out, int out_size, void* d_ws, size_t ws_size,
                              hipStream_t stream) {
    const float* x      = (const float*)d_in[0];
    const int*   src    = (const int*)  d_in[1];
    const int*   dst    = (const int*)  d_in[2];
    const float* W      = (const float*)d_in[3];
    const float* attn_l = (const float*)d_in[4];
    const float* attn_r = (const float*)d_in[5];
    const float* W_res  = (const float*)d_in[6];
    float* out = (float*)d_out;

    // workspace carve-up (floats): h | el | er | m | s | W_red  (~54.5 MB)
    float* ws   = (float*)d_ws;
    float* h    = ws;
    float* el   = h  + (size_t)N_NODES * HD;
    float* er   = el + (size_t)N_NODES * HEADS;
    float* m    = er + (size_t)N_NODES * HEADS;
    float* s    = m  + (size_t)N_NODES * HEADS;
    float* Wred = s  + (size_t)N_NODES * HEADS;

    constexpr int ROWTILES = N_NODES / 16;               // 3125
    constexpr int BLK_PER_STRIP = (ROWTILES + 7) / 8;    // 391

    // 1) fold residual projection with head-mean: W_red = 0.25 * sum_h W_res
    gat_wres_reduce<<<(IN_CH * DHEAD + 255) / 256, 256, 0, stream>>>(W_res, Wred);

    // 2) h = x @ W  (fp32 WMMA, LDS-staged B, 4 col-strips of 64)
    gat_gemm_f32_lds<HD><<<BLK_PER_STRIP * (HD / 64), 256, 0, stream>>>(x, W, h);

    // 3) out = x @ W_red  (initializes d_out with the residual/mean term)
    gat_gemm_f32_lds<DHEAD><<<BLK_PER_STRIP, 256, 0, stream>>>(x, Wred, out);

    // 4) attention logits
    gat_logits<<<(N_NODES * HEADS + 255) / 256, 256, 0, stream>>>(h, attn_l, attn_r, el, er);
    // 5) init segment max/sum
    gat_init<<<(N_NODES * HEADS + 255) / 256, 256, 0, stream>>>(m, s);
    // 6) segment max over destination neighborhoods
    gat_edge_max<<<(N_EDGES * HEADS + 255) / 256, 256, 0, stream>>>(src, dst, el, er, m);
    // 7) segment sum of exp(e - m)
    gat_edge_sum<<<(N_EDGES * HEADS + 255) / 256, 256, 0, stream>>>(src, dst, el, er, m, s);
    // 8) weighted scatter-aggregate fused with head-mean
    gat_aggregate<<<(N_EDGES * 32 + 255) / 256, 256, 0, stream>>>(src, dst, el, er, m, s, h, out);
}